// Net_40656160424624
// MI455X (gfx1250) — compile-verified
//
#include <hip/hip_runtime.h>
#include <hip/hip_bf16.h>

// PointNet++ (B=8, N=2048, K=64, M1=1024, M2=256, NC=40) for gfx1250.
// fp32 GEMMs run on V_WMMA_F32_16X16X4_F32 (CDNA5 dense fp32 matrix op).

typedef __attribute__((ext_vector_type(2))) float v2f;
typedef __attribute__((ext_vector_type(8))) float v8f;

#define K_NB 64

// ---------------------------------------------------------------------------
// Farthest point sampling, one block (256 thr) per cloud. Deterministic:
// argmax ties resolved to the lowest index, matching jnp.argmax.
// ---------------------------------------------------------------------------
__global__ void k_fps(const float* __restrict__ pos, int n, int m, int* __restrict__ idx)
{
    const int b = blockIdx.x;
    const float* p = pos + (size_t)b * n * 3;
    __shared__ float mind[2048];
    __shared__ float wv[8];
    __shared__ int   wi[8];
    __shared__ int   resi;
    const int tid  = threadIdx.x;
    const int lane = tid & 31;
    const int wav  = tid >> 5;

    const float p0x = p[0], p0y = p[1], p0z = p[2];
    for (int j = tid; j < n; j += 256) {
        float dx = p[3*j] - p0x, dy = p[3*j+1] - p0y, dz = p[3*j+2] - p0z;
        mind[j] = dx*dx + dy*dy + dz*dz;
    }
    if (tid == 0) idx[(size_t)b * m] = 0;
    __syncthreads();

    for (int s = 1; s < m; ++s) {
        float bv = -1.0f; int bi = 0;
        for (int j = tid; j < n; j += 256) {
            float v = mind[j];
            if (v > bv) { bv = v; bi = j; }
        }
        // wave32 argmax reduction (lowest index wins ties)
        for (int off = 16; off > 0; off >>= 1) {
            float ov = __shfl_down(bv, off);
            int   oi = __shfl_down(bi, off);
            if (ov > bv || (ov == bv && oi < bi)) { bv = ov; bi = oi; }
        }
        if (lane == 0) { wv[wav] = bv; wi[wav] = bi; }
        __syncthreads();
        if (tid == 0) {
            float fv = wv[0]; int fi = wi[0];
            for (int w = 1; w < 8; ++w)
                if (wv[w] > fv || (wv[w] == fv && wi[w] < fi)) { fv = wv[w]; fi = wi[w]; }
            resi = fi;
            idx[(size_t)b * m + s] = fi;
        }
        __syncthreads();
        const int sel = resi;
        const float sx = p[3*sel], sy = p[3*sel+1], sz = p[3*sel+2];
        for (int j = tid; j < n; j += 256) {
            float dx = p[3*j]-sx, dy = p[3*j+1]-sy, dz = p[3*j+2]-sz;
            float d = dx*dx + dy*dy + dz*dz;
            if (d < mind[j]) mind[j] = d;
        }
        __syncthreads();
    }
}

// ---------------------------------------------------------------------------
// Gather fps centers' positions.
// ---------------------------------------------------------------------------
__global__ void k_gather_pos(const float* __restrict__ pos, const int* __restrict__ idx,
                             int n, int m, int total, float* __restrict__ posc)
{
    int i = blockIdx.x * blockDim.x + threadIdx.x;
    if (i >= total) return;
    int b = i / m;
    int j = idx[i];
    const float* p = pos + ((size_t)b * n + j) * 3;
    posc[3*i]   = p[0];
    posc[3*i+1] = p[1];
    posc[3*i+2] = p[2];
}

// ---------------------------------------------------------------------------
// Ball query: K=64 nearest within radius^2 (iterative min-selection, lowest
// index on ties == lax.top_k semantics). One block (256 thr) per center.
// ---------------------------------------------------------------------------
__global__ void k_ballquery(const float* __restrict__ pos, const float* __restrict__ posc,
                            int n, int m, float r2,
                            int* __restrict__ nb, int* __restrict__ mask)
{
    const int ci = blockIdx.x;              // b*m + i
    const int b  = ci / m;
    const float* p = pos + (size_t)b * n * 3;
    const float* c = posc + (size_t)ci * 3;
    __shared__ float d2[2048];
    __shared__ float wv[8];
    __shared__ int   wi[8];
    __shared__ float resv;
    __shared__ int   resi;
    const int tid  = threadIdx.x;
    const int lane = tid & 31;
    const int wav  = tid >> 5;

    const float cx = c[0], cy = c[1], cz = c[2];
    for (int j = tid; j < n; j += 256) {
        float dx = p[3*j]-cx, dy = p[3*j+1]-cy, dz = p[3*j+2]-cz;
        float d = dx*dx + dy*dy + dz*dz;
        d2[j] = (d <= r2) ? d : 3.0e38f;
    }
    __syncthreads();

    const size_t base = (size_t)ci * K_NB;
    for (int k = 0; k < K_NB; ++k) {
        float bv = 3.0e38f; int bi = n;
        for (int j = tid; j < n; j += 256) {
            float v = d2[j];
            if (v < bv) { bv = v; bi = j; }
        }
        for (int off = 16; off > 0; off >>= 1) {
            float ov = __shfl_down(bv, off);
            int   oi = __shfl_down(bi, off);
            if (ov < bv || (ov == bv && oi < bi)) { bv = ov; bi = oi; }
        }
        if (lane == 0) { wv[wav] = bv; wi[wav] = bi; }
        __syncthreads();
        if (tid == 0) {
            float fv = wv[0]; int fi = wi[0];
            for (int w = 1; w < 8; ++w)
                if (wv[w] < fv || (wv[w] == fv && wi[w] < fi)) { fv = wv[w]; fi = wi[w]; }
            resv = fv; resi = fi;
            if (fv <= r2) {
                d2[fi] = 3.0e38f;                 // remove from candidate set
                nb[base + k] = fi; mask[base + k] = 1;
            }
        }
        __syncthreads();
        if (!(resv <= r2)) {                      // uniform: ball exhausted
            for (int kk = k + tid; kk < K_NB; kk += 256) { nb[base+kk] = 0; mask[base+kk] = 0; }
            break;
        }
    }
}

// ---------------------------------------------------------------------------
// Build the per-edge feature rows [x_j, pos_j - pos_c], zero-padded to lda,
// invalid (masked) rows fully zeroed.  (Zero pad cols make the GEMM K-tail
// branchless: 0 * W[clamped_k] == 0.)
// ---------------------------------------------------------------------------
__global__ void k_build_h0(const float* __restrict__ x, int C,
                           const float* __restrict__ pos, const float* __restrict__ posc,
                           const int* __restrict__ nb, const int* __restrict__ mask,
                           int n, int m, int lda, long totalRows,
                           float* __restrict__ H)
{
    long row = (long)blockIdx.x * blockDim.x + threadIdx.x;
    if (row >= totalRows) return;
    float* h = H + (size_t)row * lda;
    if (mask[row]) {
        const long ci = row >> 6;                 // center index b*m+i
        const int  b  = (int)(ci / m);
        const int  j  = nb[row];
        const float* xs = x + ((size_t)b * n + j) * C;
        for (int cc = 0; cc < C; ++cc) h[cc] = xs[cc];
        const float* pj = pos + ((size_t)b * n + j) * 3;
        const float* pc = posc + (size_t)ci * 3;
        h[C+0] = pj[0] - pc[0];
        h[C+1] = pj[1] - pc[1];
        h[C+2] = pj[2] - pc[2];
        for (int cc = C + 3; cc < lda; ++cc) h[cc] = 0.0f;
    } else {
        for (int cc = 0; cc < lda; ++cc) h[cc] = 0.0f;
    }
}

// ---------------------------------------------------------------------------
// fp32 GEMM via V_WMMA_F32_16X16X4_F32. One wave computes a 16x64 tile
// (A fragment reused across 4 column sub-tiles). Fused bias (+optional ReLU).
//
// Branchless inner loop:
//  - column indices are CLAMPED to N-1 (out-of-range cols compute garbage
//    that the guarded store never writes; matmul columns are independent),
//  - K rows are CLAMPED to Kreal-1 (A's zero pad columns annihilate them).
// A-frag layout (ISA 7.12.2, 16x4 fp32): lanes 0-15 -> K={k,k+1},
// lanes 16-31 -> K={k+2,k+3}; row = lane&15. B-frag mirrors it.
// ---------------------------------------------------------------------------
__global__ void k_gemm_wmma(const float* __restrict__ A, int lda,
                            const float* __restrict__ W, const float* __restrict__ bias,
                            float* __restrict__ C, int ldc,
                            int M, int Mvalid, int Kpad, int Kreal, int N, int relu)
{
    const int lane   = threadIdx.x & 31;
    const int wav    = threadIdx.x >> 5;
    const int tilesN = (N + 63) >> 6;
    const int tilesM = M >> 4;
    const int tile   = blockIdx.x * (blockDim.x >> 5) + wav;
    if (tile >= tilesM * tilesN) return;          // uniform per wave
    const int tm    = tile / tilesN;
    const int tn    = tile % tilesN;
    const int l15   = lane & 15;
    const int hi    = lane >> 4;                  // 0 or 1
    const int khalf = hi << 1;                    // 0 or 2
    const int c0    = tn * 64 + l15;

    // clamped (safe) column indices, loop-invariant
    const int ca = min(c0,      N - 1);
    const int cb = min(c0 + 16, N - 1);
    const int cc = min(c0 + 32, N - 1);
    const int cd = min(c0 + 48, N - 1);

    const float* Arow = A + (size_t)(tm * 16 + l15) * lda + khalf;

    v8f acc0 = {}; v8f acc1 = {}; v8f acc2 = {}; v8f acc3 = {};
    for (int k = 0; k < Kpad; k += 4) {
        v2f a;
        a.x = Arow[k];
        a.y = Arow[k + 1];
        __builtin_prefetch(Arow + k + 32, 0, 1);          // stream A ahead
        const int kr0 = min(k + khalf,     Kreal - 1);    // safe K rows
        const int kr1 = min(k + khalf + 1, Kreal - 1);
        const float* W0 = W + (size_t)kr0 * N;
        const float* W1 = W + (size_t)kr1 * N;
        v2f b0, b1, b2, b3;
        b0.x = W0[ca];  b0.y = W1[ca];
        b1.x = W0[cb];  b1.y = W1[cb];
        b2.x = W0[cc];  b2.y = W1[cc];
        b3.x = W0[cd];  b3.y = W1[cd];
        acc0 = __builtin_amdgcn_wmma_f32_16x16x4_f32(false, a, false, b0, (short)0, acc0, false, false);
        acc1 = __builtin_amdgcn_wmma_f32_16x16x4_f32(false, a, false, b1, (short)0, acc1, false, false);
        acc2 = __builtin_amdgcn_wmma_f32_16x16x4_f32(false, a, false, b2, (short)0, acc2, false, false);
        acc3 = __builtin_amdgcn_wmma_f32_16x16x4_f32(false, a, false, b3, (short)0, acc3, false, false);
    }
    #pragma unroll
    for (int v = 0; v < 8; ++v) {
        const int r = tm * 16 + v + (hi << 3);    // C/D layout: VGPR v -> M = v (+8 for hi lanes)
        if (r >= Mvalid) continue;
        float vals[4] = { acc0[v], acc1[v], acc2[v], acc3[v] };
        #pragma unroll
        for (int sub = 0; sub < 4; ++sub) {
            const int col = c0 + sub * 16;
            if (col < N) {
                float o = vals[sub] + bias[col];
                if (relu) o = fmaxf(o, 0.0f);
                C[(size_t)r * ldc + col] = o;
            }
        }
    }
}

// ---------------------------------------------------------------------------
// Masked BatchNorm statistics: deterministic two-stage reduction.
// ---------------------------------------------------------------------------
__global__ void k_bn_partial(const float* __restrict__ H, const int* __restrict__ mask,
                             int rows, int N, int rowsPerBlk, int stride,
                             float* __restrict__ part)
{
    __shared__ float s[1024];
    __shared__ float s2[1024];
    const int tid = threadIdx.x;
    for (int c = tid; c < N; c += blockDim.x) { s[c] = 0.0f; s2[c] = 0.0f; }
    __syncthreads();
    const int r0 = blockIdx.x * rowsPerBlk;
    const int r1 = min(r0 + rowsPerBlk, rows);
    int cnt = 0;
    for (int r = r0; r < r1; ++r) {
        const bool valid = (mask == nullptr) || (mask[r] != 0);
        if (valid) {
            ++cnt;
            const float* h = H + (size_t)r * N;
            for (int c = tid; c < N; c += blockDim.x) {
                float v = h[c]; s[c] += v; s2[c] += v * v;
            }
        }
    }
    __syncthreads();
    float* o = part + (size_t)blockIdx.x * stride;
    for (int c = tid; c < N; c += blockDim.x) { o[c] = s[c]; o[N + c] = s2[c]; }
    if (tid == 0) o[2 * N] = (float)cnt;
}

__global__ void k_bn_finalize(const float* __restrict__ part, int nblk, int N, int stride,
                              float* __restrict__ sum, float* __restrict__ sumsq,
                              float* __restrict__ cnt)
{
    const int c = blockIdx.x * blockDim.x + threadIdx.x;
    if (c < N) {
        float s = 0.0f, s2 = 0.0f;
        for (int b = 0; b < nblk; ++b) {
            s  += part[(size_t)b * stride + c];
            s2 += part[(size_t)b * stride + N + c];
        }
        sum[c] = s; sumsq[c] = s2;
    }
    if (c == 0) {
        float n = 0.0f;
        for (int b = 0; b < nblk; ++b) n += part[(size_t)b * stride + 2 * N];
        cnt[0] = n;
    }
}

__global__ void k_bn_apply(float* __restrict__ H, const int* __restrict__ mask,
                           long total, int N,
                           const float* __restrict__ sum, const float* __restrict__ sumsq,
                           const float* __restrict__ cnt,
                           const float* __restrict__ g, const float* __restrict__ bt)
{
    long i = (long)blockIdx.x * blockDim.x + threadIdx.x;
    if (i >= total) return;
    long r = i / N; int c = (int)(i % N);
    if (mask != nullptr && mask[r] == 0) { H[i] = 0.0f; return; }
    float n    = fmaxf(cnt[0], 1.0f);
    float mean = sum[c] / n;
    float var  = fmaxf(sumsq[c] / n - mean * mean, 0.0f);
    H[i] = g[c] * (H[i] - mean) * rsqrtf(var + 1e-5f) + bt[c];
}

// ---------------------------------------------------------------------------
// Masked max over K neighbors (invalid -> 0, matching isfinite fixup).
// ---------------------------------------------------------------------------
__global__ void k_maxpool(const float* __restrict__ H, const int* __restrict__ mask,
                          int centers, int N, float* __restrict__ out)
{
    int i = blockIdx.x * blockDim.x + threadIdx.x;
    if (i >= centers * N) return;
    int ci = i / N, c = i % N;
    const float* h  = H + ((size_t)ci * K_NB) * N + c;
    const int*   mk = mask + (size_t)ci * K_NB;
    float best = -3.0e38f; int any = 0;
    for (int k = 0; k < K_NB; ++k) {
        if (mk[k]) { float v = h[(size_t)k * N]; if (v > best) best = v; any = 1; }
    }
    out[i] = any ? best : 0.0f;
}

// Global max pool over m centers; rows >= Bvalid zero-filled (pad to 16).
__global__ void k_globalmax(const float* __restrict__ H, int m, int N, int Bvalid, int Mpad,
                            float* __restrict__ out)
{
    int i = blockIdx.x * blockDim.x + threadIdx.x;
    if (i >= Mpad * N) return;
    int b = i / N, c = i % N;
    if (b >= Bvalid) { out[i] = 0.0f; return; }
    float best = -3.0e38f;
    for (int j = 0; j < m; ++j) {
        float v = H[((size_t)b * m + j) * N + c];
        if (v > best) best = v;
    }
    out[i] = best;
}

// SA3 input: concat(x2, pos2), zero-padded to lda.
__global__ void k_concat(const float* __restrict__ xf, const float* __restrict__ pf,
                         int total, int C, int lda, float* __restrict__ H)
{
    int i = blockIdx.x * blockDim.x + threadIdx.x;
    if (i >= total) return;
    int r = i / lda, c = i % lda;
    float v;
    if (c < C)          v = xf[(size_t)r * C + c];
    else if (c < C + 3) v = pf[(size_t)r * 3 + (c - C)];
    else                v = 0.0f;
    H[i] = v;
}

// ---------------------------------------------------------------------------
// Host orchestration
// ---------------------------------------------------------------------------
static void mlp_layer(const float* A, int lda, const float* W, const float* bias,
                      const float* g, const float* bt,
                      float* Cmat, int rows, int Kpad, int Kreal, int N,
                      const int* mask,
                      float* part, float* bsum, float* bsq, float* bcnt,
                      hipStream_t stream)
{
    const int tiles  = (rows / 16) * ((N + 63) / 64);
    const int blocks = (tiles + 3) / 4;
    k_gemm_wmma<<<blocks, 128, 0, stream>>>(A, lda, W, bias, Cmat, N, rows, rows, Kpad, Kreal, N, 1);
    const int rpb    = 2048;
    const int nblk   = (rows + rpb - 1) / rpb;
    const int stride = 2 * 1024 + 8;
    k_bn_partial<<<nblk, 256, 0, stream>>>(Cmat, mask, rows, N, rpb, stride, part);
    k_bn_finalize<<<(N + 255) / 256, 256, 0, stream>>>(part, nblk, N, stride, bsum, bsq, bcnt);
    long total = (long)rows * N;
    k_bn_apply<<<(unsigned)((total + 255) / 256), 256, 0, stream>>>(Cmat, mask, total, N,
                                                                    bsum, bsq, bcnt, g, bt);
}

extern "C" void kernel_launch(void* const* d_in, const int* in_sizes, int n_in,
                              void* d_out, int out_size, void* d_ws, size_t ws_size,
                              hipStream_t stream)
{
    (void)in_sizes; (void)n_in; (void)out_size; (void)ws_size;
    const float* x   = (const float*)d_in[0];
    const float* pos = (const float*)d_in[1];
    auto F = [&](int i) { return (const float*)d_in[i]; };
    // param leaf order: sa1(3..14), sa2(15..26), sa3(27..38), head(39..44)

    char* ws = (char*)d_ws;
    size_t off = 0;
    auto alloc = [&](size_t bytes) -> void* {
        void* p = (void*)(ws + off);
        off += (bytes + 255) & ~(size_t)255;
        return p;
    };
    float* bufA = (float*)alloc((size_t)524288 * 64  * sizeof(float));   // 134 MB
    float* bufB = (float*)alloc((size_t)524288 * 128 * sizeof(float));   // 268 MB
    int*   idxb = (int*)  alloc((size_t)8192 * sizeof(int));
    float* pos1 = (float*)alloc((size_t)8192 * 3 * sizeof(float));
    float* pos2 = (float*)alloc((size_t)2048 * 3 * sizeof(float));
    int*   nb   = (int*)  alloc((size_t)524288 * sizeof(int));
    int*   mask = (int*)  alloc((size_t)524288 * sizeof(int));
    float* x1   = (float*)alloc((size_t)8192 * 128 * sizeof(float));
    float* x2   = (float*)alloc((size_t)2048 * 256 * sizeof(float));
    float* part = (float*)alloc((size_t)256 * (2 * 1024 + 8) * sizeof(float));
    float* bsum = (float*)alloc((size_t)1024 * sizeof(float));
    float* bsq  = (float*)alloc((size_t)1024 * sizeof(float));
    float* bcnt = (float*)alloc((size_t)256);
    float* gpad = (float*)alloc((size_t)16 * 1024 * sizeof(float));
    float* h1   = (float*)alloc((size_t)16 * 512 * sizeof(float));
    float* h2   = (float*)alloc((size_t)16 * 256 * sizeof(float));

    // ---------------- SA1: n=2048 -> m=1024, r^2=0.04, channels 6->64->64->128
    k_fps<<<8, 256, 0, stream>>>(pos, 2048, 1024, idxb);
    k_gather_pos<<<(8192 + 255) / 256, 256, 0, stream>>>(pos, idxb, 2048, 1024, 8192, pos1);
    k_ballquery<<<8 * 1024, 256, 0, stream>>>(pos, pos1, 2048, 1024, 0.04f, nb, mask);
    k_build_h0<<<(524288 + 255) / 256, 256, 0, stream>>>(x, 3, pos, pos1, nb, mask,
                                                         2048, 1024, 8, 524288L, bufA);
    mlp_layer(bufA, 8,  F(3),  F(4),  F(5),  F(6),  bufB, 524288, 8,  6,  64,  mask, part, bsum, bsq, bcnt, stream);
    mlp_layer(bufB, 64, F(7),  F(8),  F(9),  F(10), bufA, 524288, 64, 64, 64,  mask, part, bsum, bsq, bcnt, stream);
    mlp_layer(bufA, 64, F(11), F(12), F(13), F(14), bufB, 524288, 64, 64, 128, mask, part, bsum, bsq, bcnt, stream);
    k_maxpool<<<(8192 * 128 + 255) / 256, 256, 0, stream>>>(bufB, mask, 8192, 128, x1);

    // ---------------- SA2: n=1024 -> m=256, r^2=0.16, channels 131->128->128->256
    k_fps<<<8, 256, 0, stream>>>(pos1, 1024, 256, idxb);
    k_gather_pos<<<(2048 + 255) / 256, 256, 0, stream>>>(pos1, idxb, 1024, 256, 2048, pos2);
    k_ballquery<<<8 * 256, 256, 0, stream>>>(pos1, pos2, 1024, 256, 0.16f, nb, mask);
    k_build_h0<<<(131072 + 255) / 256, 256, 0, stream>>>(x1, 128, pos1, pos2, nb, mask,
                                                         1024, 256, 132, 131072L, bufA);
    mlp_layer(bufA, 132, F(15), F(16), F(17), F(18), bufB, 131072, 132, 131, 128, mask, part, bsum, bsq, bcnt, stream);
    mlp_layer(bufB, 128, F(19), F(20), F(21), F(22), bufA, 131072, 128, 128, 128, mask, part, bsum, bsq, bcnt, stream);
    mlp_layer(bufA, 128, F(23), F(24), F(25), F(26), bufB, 131072, 128, 128, 256, mask, part, bsum, bsq, bcnt, stream);
    k_maxpool<<<(2048 * 256 + 255) / 256, 256, 0, stream>>>(bufB, mask, 2048, 256, x2);

    // ---------------- SA3: rows=2048 (no mask), channels 259->256->512->1024
    k_concat<<<(2048 * 260 + 255) / 256, 256, 0, stream>>>(x2, pos2, 2048 * 260, 256, 260, bufA);
    mlp_layer(bufA, 260, F(27), F(28), F(29), F(30), bufB, 2048, 260, 259, 256,  nullptr, part, bsum, bsq, bcnt, stream);
    mlp_layer(bufB, 256, F(31), F(32), F(33), F(34), bufA, 2048, 256, 256, 512,  nullptr, part, bsum, bsq, bcnt, stream);
    mlp_layer(bufA, 512, F(35), F(36), F(37), F(38), bufB, 2048, 512, 512, 1024, nullptr, part, bsum, bsq, bcnt, stream);
    k_globalmax<<<(16 * 1024 + 255) / 256, 256, 0, stream>>>(bufB, 256, 1024, 8, 16, gpad);

    // ---------------- Head: 1024->512->256->40 (rows padded 8 -> 16)
    auto gemm = [&](const float* A, int lda, const float* W, const float* b, float* C, int ldc,
                    int M, int Mv, int Kp, int Kr, int Ncol, int relu) {
        int tiles = (M / 16) * ((Ncol + 63) / 64);
        k_gemm_wmma<<<(tiles + 3) / 4, 128, 0, stream>>>(A, lda, W, b, C, ldc, M, Mv, Kp, Kr, Ncol, relu);
    };
    gemm(gpad, 1024, F(39), F(40), h1, 512, 16, 16, 1024, 1024, 512, 1);
    gemm(h1,   512,  F(41), F(42), h2, 256, 16, 16, 512,  512,  256, 1);
    gemm(h2,   256,  F(43), F(44), (float*)d_out, 40, 16, 8, 256, 256, 40, 0);
}